// SSMLayer_46084999086636
// MI455X (gfx1250) — compile-verified
//
#include <hip/hip_runtime.h>

typedef __attribute__((ext_vector_type(2))) float v2f;
typedef __attribute__((ext_vector_type(4))) float v4f;
typedef __attribute__((ext_vector_type(8))) float v8f;

#define NBATCH 8
#define NFEAT  256
#define HD     64
#define NT     1024
#define AUGC   130          // 129 augmented cols, padded -> conflict-free
#define XSTR   68           // 68 dwords = 272B: conflict-free AND 16B-aligned rows
#define YCHUNK 256

__launch_bounds__(128)
__global__ void ssm_scan_wmma_f32(const float* __restrict__ x,
                                  const float* __restrict__ Amat,
                                  const float* __restrict__ Bmat,
                                  const float* __restrict__ Cmat,
                                  float* __restrict__ out)
{
    // Phase-1 Gauss-Jordan tableau (64*130 = 8320 floats) aliases the
    // phase-2 u (8192) + y (2048) staging region.
    __shared__ float smem[NBATCH*NT + NBATCH*YCHUNK + 16*XSTR + 128];
    float* aug   = smem;                                  // phase 1
    float* u_lds = smem;                                  // phase 2: [t][b]
    float* y_lds = smem + NBATCH*NT;                      // [i][b], 256-step chunk
    float* X_lds = smem + NBATCH*NT + NBATCH*YCHUNK;      // state, [col][row] stride 68
    float* p_lds = X_lds + 16*XSTR;                       // 8x16 partial-y slots

    const int f   = blockIdx.x;
    const int tid = threadIdx.x;

    // -------- Phase 1: bilinear discretization, M X = [I + A/2 | B] --------
    const float* Af = Amat + f*HD*HD;
    for (int e = tid; e < HD*HD; e += 128) {
        int r = e >> 6, c = e & 63;
        float a   = Af[e];
        float eye = (r == c) ? 1.0f : 0.0f;
        aug[r*AUGC + c]      = eye - 0.5f*a;   // M = I - A/2
        aug[r*AUGC + 64 + c] = eye + 0.5f*a;   // RHS block for A_bar
    }
    for (int r = tid; r < HD; r += 128)
        aug[r*AUGC + 128] = Bmat[f*HD + r];    // RHS col for B_bar (STEP = 1)
    __syncthreads();

    {   // Gauss-Jordan, no pivoting (M = 1.75 I - G/2 is diagonally dominant).
        // Each row handled by 2 threads (column halves).
        const int row  = tid & 63;
        const int c0   = (tid >> 6) ? 64 : 0;
        const int ncol = (tid >> 6) ? 65 : 64;
        for (int p = 0; p < HD; ++p) {
            float fac = aug[row*AUGC + p] / aug[p*AUGC + p];
            __syncthreads();                   // reads done before overwrites
            if (row != p) {
                for (int c = 0; c < ncol; ++c)
                    aug[row*AUGC + c0 + c] -= fac * aug[p*AUGC + c0 + c];
            }
            __syncthreads();
        }
    }

    // -------- Extract WMMA A-operands (A_bar) + Bv, Cv into registers --------
    const int wv   = tid >> 5;        // wave id = M-tile (rows 16wv..16wv+15)
    const int ln   = tid & 31;
    const int hi   = ln >> 4;         // lane half
    const int col  = ln & 15;         // batch column / matrix N index
    const int koff = hi << 1;         // A-op: lanes 16-31 hold K = 4k+2,4k+3
    const int arow = wv*16 + col;     // A-operand M row for this lane
    const int row2 = wv*16 + hi*8;    // first state row in D accumulator

    v2f a_op[16];
    {
        float rinv = 1.0f / aug[arow*AUGC + arow];
        #pragma unroll
        for (int k = 0; k < 16; ++k) {
            a_op[k].x = aug[arow*AUGC + 64 + 4*k + koff]     * rinv;
            a_op[k].y = aug[arow*AUGC + 64 + 4*k + koff + 1] * rinv;
        }
    }
    float bvr[8], cvr[8];
    #pragma unroll
    for (int r = 0; r < 8; ++r) {
        int rr = row2 + r;
        bvr[r] = aug[rr*AUGC + 128] / aug[rr*AUGC + rr];
        cvr[r] = Cmat[f*HD + rr];
    }
    __syncthreads();                  // tableau dead; region becomes u/y staging

    // -------- Phase 2 setup: stage u transposed, zero state --------
    for (int e = tid; e < NBATCH*NT; e += 128) {
        int b = e >> 10, i = e & (NT - 1);
        u_lds[i*NBATCH + b] = x[(size_t)(b*NFEAT + f)*NT + i];
    }
    for (int e = tid; e < 16*XSTR; e += 128) X_lds[e] = 0.0f;
    __syncthreads();

    // -------- Sequential scan: X = A_bar*X + Bv u_t ; y = Cv . X --------
    for (int t = 0; t < NT; ++t) {
        v8f D = {};
        #pragma unroll
        for (int k = 0; k < 16; ++k) {
            // B-operand (4x16 slice of state): rows 4k+koff, 4k+koff+1 -> ds_load_b64
            v2f bop = *(const v2f*)&X_lds[col*XSTR + 4*k + koff];
            D = __builtin_amdgcn_wmma_f32_16x16x4_f32(false, a_op[k], false, bop,
                                                      (short)0, D, false, false);
        }

        float u = (col < NBATCH) ? u_lds[t*NBATCH + col] : 0.0f;
        float psum = 0.0f;
        #pragma unroll
        for (int r = 0; r < 8; ++r) {
            D[r] += bvr[r] * u;       // rank-1 input injection
            psum += cvr[r] * D[r];    // partial Cv . x over this lane's 8 rows
        }
        p_lds[(wv*2 + hi)*16 + col] = psum;
        __syncthreads();              // all state reads + p writes complete

        // State write-back: 8 contiguous floats per lane -> 2x ds_store_b128
        v4f lo4 = { D[0], D[1], D[2], D[3] };
        v4f hi4 = { D[4], D[5], D[6], D[7] };
        *(v4f*)&X_lds[col*XSTR + row2]     = lo4;
        *(v4f*)&X_lds[col*XSTR + row2 + 4] = hi4;

        if (tid < NBATCH) {           // reduce 8 partials -> y[b], stage in LDS
            float y = 0.0f;
            #pragma unroll
            for (int j = 0; j < 8; ++j) y += p_lds[j*16 + tid];
            y_lds[(t & (YCHUNK-1))*NBATCH + tid] = y;
        }

        if ((t & (YCHUNK-1)) == (YCHUNK-1)) {   // coalesced chunk flush
            __syncthreads();
            int tbase = t & ~(YCHUNK-1);
            for (int e = tid; e < NBATCH*YCHUNK; e += 128) {
                int b = e >> 8, i = e & (YCHUNK-1);
                out[(size_t)(b*NFEAT + f)*NT + tbase + i] = y_lds[i*NBATCH + b];
            }
        }
        __syncthreads();              // writes visible before next step's reads
    }
}

extern "C" void kernel_launch(void* const* d_in, const int* in_sizes, int n_in,
                              void* d_out, int out_size, void* d_ws, size_t ws_size,
                              hipStream_t stream) {
    (void)in_sizes; (void)n_in; (void)d_ws; (void)ws_size; (void)out_size;
    const float* x = (const float*)d_in[0];   // (8, 256, 1024)
    const float* A = (const float*)d_in[1];   // (256, 64, 64)
    const float* B = (const float*)d_in[2];   // (256, 64, 1)
    const float* C = (const float*)d_in[3];   // (256, 1, 64)
    float* out = (float*)d_out;               // (8, 256, 1024)
    ssm_scan_wmma_f32<<<dim3(NFEAT), dim3(128), 0, stream>>>(x, A, B, C, out);
}